// VoxelResSPSAttnQuantiseizer_39479339385417
// MI455X (gfx1250) — compile-verified
//
#include <hip/hip_runtime.h>
#include <cstdint>

#define N_VOX  8192
#define M_KEY  8192
#define KEEP   4096
#define HID    32
#define NEGV   (-1000000000.0f)
#define MT     1024                 // keys per LDS chunk (16 KB)
#define NCHUNK (M_KEY / MT)
#define L2E    1.4426950408889634f  // log2(e)

typedef __attribute__((ext_vector_type(2))) float v2f;
typedef __attribute__((ext_vector_type(8))) float v8f;

// key record: x=batch, y=z, z=y, w=x  (matches (b,z,y,x) memory order)
struct alignas(16) F4 { float x, y, z, w; };

typedef __attribute__((address_space(3))) F4 lds_f4;

__device__ __forceinline__ void async_load_b128(lds_f4* dst, const F4* src) {
  // gfx1250 async DMA: global -> LDS, tracked by ASYNCcnt
  asm volatile("global_load_async_to_lds_b128 %0, %1, off"
               :: "v"(dst), "v"(src) : "memory");
}

// lane-pair exchange (lanes l <-> l^16): ds_swizzle SWAPX16, immediate pattern
__device__ __forceinline__ float swap16(float x) {
  return __int_as_float(__builtin_amdgcn_ds_swizzle(__float_as_int(x), 0x401f));
}

// native 2^x (v_exp_f32)
__device__ __forceinline__ float fexp2(float x) {
  return __builtin_amdgcn_exp2f(x);
}

// ---------------------------------------------------------------------------
// Kernel A: fused masked attention (flash) + 3->32->1 MLP -> importance[N]
// One wave = 16 queries; block = 8 waves = 128 queries; grid = 64 blocks.
// ---------------------------------------------------------------------------
__global__ __launch_bounds__(256) void imp_kernel(
    const int*   __restrict__ vcoords,   // N x 4 (b,z,y,x) int32
    const F4*    __restrict__ keys,      // M x 4 (b,z,y,x) f32
    const float* __restrict__ w1, const float* __restrict__ b1,
    const float* __restrict__ w2, const float* __restrict__ b2,
    float*       __restrict__ imp_out,   // d_out region 3 (N floats)
    float*       __restrict__ imp_ws)    // workspace copy (N floats)
{
  __shared__ F4 kbuf[2][MT];             // 32 KB double buffer

  const int  tid  = threadIdx.x;
  const int  lane = tid & 31;            // wave32
  const int  wave = tid >> 5;
  const bool hi   = (lane >= 16);
  const int  n    = lane & 15;           // query column owned by this lane

  const int qi = blockIdx.x * 128 + wave * 16 + n;

  // Query (B operand, 4x16): k0=z, k1=y, k2=x, k3=0; fold in 1/sqrt(3).
  const int*  qc = vcoords + qi * 4;
  const float qb = (float)qc[0];
  const float qz = (float)qc[1];
  const float qy = (float)qc[2];
  const float qx = (float)qc[3];
  const float r3 = 0.57735026918962576f;
  v2f bq;
  bq.x = (hi ? qx : qz) * r3;
  bq.y = hi ? 0.0f : (qy * r3);

  // Online-softmax running state (per lane: query n, this lane's 8 keys/tile)
  float rm = NEGV, rs = 0.0f, o0 = 0.0f, o1 = 0.0f, o2 = 0.0f;

  // Prefetch chunk 0 (4 x b128 per thread = 16 KB per block)
  {
    lds_f4* db = (lds_f4*)&kbuf[0][0];
    #pragma unroll
    for (int r = 0; r < 4; ++r) {
      const int idx = tid + 256 * r;
      async_load_b128(db + idx, keys + idx);
    }
  }

  for (int c = 0; c < NCHUNK; ++c) {
    const int buf = c & 1;
    if (c + 1 < NCHUNK) {
      lds_f4*   db = (lds_f4*)&kbuf[buf ^ 1][0];
      const F4* gp = keys + (c + 1) * MT;
      #pragma unroll
      for (int r = 0; r < 4; ++r) {
        const int idx = tid + 256 * r;
        async_load_b128(db + idx, gp + idx);
      }
      // in-order ASYNCcnt: <=4 outstanding => chunk c's 4 loads done
      asm volatile("s_wait_asynccnt 0x4" ::: "memory");
    } else {
      asm volatile("s_wait_asynccnt 0x0" ::: "memory");
    }
    __syncthreads();

    const F4* kb = &kbuf[buf][0];
    for (int t = 0; t < MT / 16; ++t) {
      const int kt = t * 16;

      // A operand (16 keys x 4): lanes 0-15 -> (k0,k1)=(z,y); 16-31 -> (k2,k3)=(x,0)
      const F4 ka = kb[kt + n];
      v2f a;
      a.x = hi ? ka.w : ka.y;
      a.y = hi ? 0.0f : ka.z;

      // scores[key m][query n] for a 16x16 tile, one instruction
      v8f cacc = {};
      cacc = __builtin_amdgcn_wmma_f32_16x16x4_f32(
          false, a, false, bq, (short)0, cacc, false, false);

      // This lane: query n, keys m = kt + j + (hi?8:0), scores in cacc[j]
      const int kofs = kt + (hi ? 8 : 0);
      float sj[8];
      float tmax = NEGV;
      #pragma unroll
      for (int j = 0; j < 8; ++j) {
        const F4 kk = kb[kofs + j];               // LDS broadcast read
        const float s = (kk.x == qb) ? cacc[j] : NEGV;
        sj[j] = s;
        tmax  = fmaxf(tmax, s);
      }
      tmax = fmaxf(tmax, swap16(tmax));           // lane-pair shares tile max
      const float nm  = fmaxf(rm, tmax);
      const float nmB = nm * L2E;                 // shift+scale folded once
      const float sc  = fexp2(fmaf(rm, L2E, -nmB));
      rs *= sc; o0 *= sc; o1 *= sc; o2 *= sc;
      #pragma unroll
      for (int j = 0; j < 8; ++j) {
        const F4 kk = kb[kofs + j];
        const float p = fexp2(fmaf(sj[j], L2E, -nmB));  // 1 FMA + 1 TRANS
        rs += p;
        o0 = fmaf(p, kk.y, o0);   // z
        o1 = fmaf(p, kk.z, o1);   // y
        o2 = fmaf(p, kk.w, o2);   // x
      }
      rm = nm;
    }
    __syncthreads();
  }

  // Combine lane pair (identical rm on both halves)
  rs += swap16(rs);
  o0 += swap16(o0);
  o1 += swap16(o1);
  o2 += swap16(o2);

  const float inv = 1.0f / rs;
  const float c0 = o0 * inv, c1 = o1 * inv, c2 = o2 * inv;

  // MLP: h = relu(cw @ w1 + b1); imp = sigmoid(h @ w2 + b2)
  float acc = b2[0];
  #pragma unroll 8
  for (int j = 0; j < HID; ++j) {
    float h = fmaf(c0, w1[j],
              fmaf(c1, w1[HID + j],
              fmaf(c2, w1[2 * HID + j], b1[j])));
    h   = fmaxf(h, 0.0f);
    acc = fmaf(h, w2[j], acc);
  }
  const float imp = 1.0f / (1.0f + fexp2(-acc * L2E));

  if (!hi) {
    imp_out[qi] = imp;
    imp_ws[qi]  = imp;
  }
}

// ---------------------------------------------------------------------------
// Kernel B: exact stable rank (counting) + top-half gather.
// rank_i = #{v_j < v_i} + #{v_j == v_i && j < i}; keep rank >= KEEP at
// output slot rank - KEEP  ==  reference's order[N/2:] ordering.
// ---------------------------------------------------------------------------
__global__ __launch_bounds__(256) void rank_kernel(
    const float* __restrict__ imp,       // N (workspace)
    const int*   __restrict__ vcoords,   // N x 4 int32
    const float* __restrict__ voxels,    // N x 5 x 4 f32
    float*       __restrict__ out)       // d_out base
{
  __shared__ float simp[N_VOX];          // 32 KB
  const int tid = threadIdx.x;
  for (int j = tid; j < N_VOX; j += 256) simp[j] = imp[j];
  __syncthreads();

  const int   i = blockIdx.x * 256 + tid;
  const float v = simp[i];
  int cnt = 0;
  const F4* sp = (const F4*)simp;        // b128 LDS reads
  for (int j4 = 0; j4 < N_VOX / 4; ++j4) {
    const F4 q = sp[j4];
    const int j = j4 * 4;
    cnt += (q.x < v) || (q.x == v && (j + 0) < i);
    cnt += (q.y < v) || (q.y == v && (j + 1) < i);
    cnt += (q.z < v) || (q.z == v && (j + 2) < i);
    cnt += (q.w < v) || (q.w == v && (j + 3) < i);
  }

  if (cnt >= KEEP) {
    const int pos = cnt - KEEP;
    float* out_coords = out;                         // 4096*4
    float* out_voxels = out + KEEP * 4;              // 4096*20
    float* out_kimp   = out + KEEP * 4 + KEEP * 20;  // 4096
    const int* qc = vcoords + i * 4;
    #pragma unroll
    for (int c = 0; c < 4; ++c) out_coords[pos * 4 + c] = (float)qc[c];
    const float* vx = voxels + i * 20;
    #pragma unroll
    for (int t = 0; t < 20; ++t) out_voxels[pos * 20 + t] = vx[t];
    out_kimp[pos] = v;
  }
}

// ---------------------------------------------------------------------------
extern "C" void kernel_launch(void* const* d_in, const int* in_sizes, int n_in,
                              void* d_out, int out_size, void* d_ws, size_t ws_size,
                              hipStream_t stream) {
  (void)in_sizes; (void)n_in; (void)out_size; (void)ws_size;
  const int*   vcoords = (const int*)  d_in[0];
  const F4*    keys    = (const F4*)   d_in[1];
  const float* voxels  = (const float*)d_in[2];
  const float* w1      = (const float*)d_in[3];
  const float* b1      = (const float*)d_in[4];
  const float* w2      = (const float*)d_in[5];
  const float* b2      = (const float*)d_in[6];

  float* out      = (float*)d_out;
  float* imp_full = out + KEEP * 4 + KEEP * 20 + KEEP;  // region 3: importance[N]
  float* ws       = (float*)d_ws;

  imp_kernel<<<dim3(N_VOX / 128), dim3(256), 0, stream>>>(
      vcoords, keys, w1, b1, w2, b2, imp_full, ws);
  rank_kernel<<<dim3(N_VOX / 256), dim3(256), 0, stream>>>(
      ws, vcoords, voxels, out);
}